// KNN_4037269258633
// MI455X (gfx1250) — compile-verified
//
#include <hip/hip_runtime.h>

// KNN (B=4, N=M=8192, C=3, k=16) for gfx1250.
// Strategy: V_WMMA_F32_16X16X4_F32 computes 16x16 tiles of the selection key
//   key[m][q] = ||x2_m||^2 - 2 * <x2_m, x1_q>
// directly:  A = (-2 * xyz2 tile) [16x4, K padded with 0],
//            B = xyz1^T tile      [4x16],
//            C = ||x2||^2 broadcast per row.
// Lanes own queries (D columns); per-lane sorted top-16 register list with a
// guarded unrolled insert chain; lane pair (j, j+16) merged via shfl_xor+bitonic.

typedef float v2f __attribute__((ext_vector_type(2)));
typedef float v8f __attribute__((ext_vector_type(8)));

constexpr int B_ = 4;
constexpr int N_ = 8192;
constexpr int M_ = 8192;
constexpr int K_ = 16;     // top-k
constexpr int CHUNK = 1024; // xyz2 points staged per LDS buffer
constexpr int QPB = 128;    // queries per block (8 waves * 16)

__global__ __launch_bounds__(256) void knn_wmma_kernel(
    const float* __restrict__ xyz1, const float* __restrict__ xyz2,
    float* __restrict__ out_dist, int* __restrict__ out_idx) {
  __shared__ __align__(16) float lds[2][CHUNK * 4];

  const int tid  = threadIdx.x;
  const int lane = tid & 31;
  const int wave = tid >> 5;
  const int b    = blockIdx.y;
  const int qbase = blockIdx.x * QPB + wave * 16;
  const int lo   = lane & 15;
  const bool hi  = lane >= 16;
  const int hi8  = hi ? 8 : 0;

  // ---- B operand (queries), loop-invariant. 4x16 layout:
  // lanes 0-15: K=0 (x), K=1 (y); lanes 16-31: K=2 (z), K=3 (0).
  const float* p1 = xyz1 + ((size_t)b * N_ + qbase + lo) * 3;
  const float qx = p1[0], qy = p1[1], qz = p1[2];
  const float n1 = qx * qx + qy * qy + qz * qz;
  v2f bmat;
  bmat[0] = hi ? qz : qx;
  bmat[1] = hi ? 0.0f : qy;

  // ---- per-lane sorted top-16 (ascending; tk[15] = current worst)
  float tk[K_];
  int   ti[K_];
#pragma unroll
  for (int i = 0; i < K_; i++) { tk[i] = __builtin_inff(); ti[i] = -1; }

  // ---- LDS staging: store (-2x, -2y, -2z, x^2+y^2+z^2) per xyz2 point
  auto stage = [&](int c, int bufi) {
    const float* src = xyz2 + ((size_t)b * M_ + (size_t)c * CHUNK) * 3;
#pragma unroll
    for (int p = 0; p < CHUNK / 256; p++) {
      const int pt = tid + p * 256;
      const float x = src[pt * 3 + 0];
      const float y = src[pt * 3 + 1];
      const float z = src[pt * 3 + 2];
      float4 v = make_float4(-2.0f * x, -2.0f * y, -2.0f * z,
                             x * x + y * y + z * z);
      *(float4*)&lds[bufi][pt * 4] = v;
    }
  };

  stage(0, 0);
  __syncthreads();

  constexpr int NCH = M_ / CHUNK;
  for (int c = 0; c < NCH; c++) {
    const int bufi = c & 1;
    if (c + 1 < NCH) stage(c + 1, bufi ^ 1);  // overlap next-chunk fill

    const float* buf = lds[bufi];
    for (int t = 0; t < CHUNK / 16; t++) {
      // A operand: 16x4, lanes 0-15: K=0,1 of point (t*16+lo);
      //            lanes 16-31: K=2,3 (z, 0-pad) of same point.
      const float4 p = *(const float4*)&buf[(t * 16 + lo) * 4];
      v2f amat;
      amat[0] = hi ? p.z : p.x;
      amat[1] = hi ? 0.0f : p.y;

      // C operand: row-constant ||x2||^2 (broadcast LDS reads)
      v8f cmat;
#pragma unroll
      for (int r = 0; r < 8; r++)
        cmat[r] = buf[(t * 16 + r + hi8) * 4 + 3];

      // D = A*B + C = ||x2||^2 - 2<x2,x1>  (the selection key)
      v8f acc = __builtin_amdgcn_wmma_f32_16x16x4_f32(
          false, amat, false, bmat, (short)0, cmat, false, false);

      const int mbase = c * CHUNK + t * 16 + hi8;
#pragma unroll
      for (int r = 0; r < 8; r++) {
        const float key = acc[r];
        if (key < tk[K_ - 1]) {  // guarded insert (rarely taken)
          float cv = key;
          int   ci = mbase + r;
#pragma unroll
          for (int i = 0; i < K_; i++) {
            const bool cl = cv < tk[i];
            const float nt = cl ? cv : tk[i];
            const int  nti = cl ? ci : ti[i];
            cv = cl ? tk[i] : cv;
            ci = cl ? ti[i] : ci;
            tk[i] = nt;
            ti[i] = nti;
          }
        }
      }
    }
    __syncthreads();
  }

  // ---- merge lane j with lane j+16 (each holds sorted 16-lists)
  float ok[K_];
  int   oi[K_];
#pragma unroll
  for (int i = 0; i < K_; i++) {
    ok[i] = __shfl_xor(tk[i], 16, 32);
    oi[i] = __shfl_xor(ti[i], 16, 32);
  }
  // bitonic min-pair: lower 16 of the 32-element union (bitonic sequence)
  float sk[K_];
  int   si[K_];
#pragma unroll
  for (int i = 0; i < K_; i++) {
    const float ov = ok[K_ - 1 - i];
    const int  ovi = oi[K_ - 1 - i];
    const bool cl  = ov < tk[i];
    sk[i] = cl ? ov : tk[i];
    si[i] = cl ? ovi : ti[i];
  }
  // bitonic clean: sort the 16 ascending (4 stages, static network)
#pragma unroll
  for (int d = 8; d >= 1; d >>= 1) {
#pragma unroll
    for (int i = 0; i < K_; i++) {
      if ((i & d) == 0) {
        const int j = i + d;
        const bool cl = sk[j] < sk[i];
        const float a0 = cl ? sk[j] : sk[i];
        const float a1 = cl ? sk[i] : sk[j];
        const int  b0 = cl ? si[j] : si[i];
        const int  b1 = cl ? si[i] : si[j];
        sk[i] = a0; sk[j] = a1; si[i] = b0; si[j] = b1;
      }
    }
  }

  // ---- low lanes own the queries: add ||x1||^2, sqrt, store
  if (!hi) {
    const size_t base = ((size_t)b * N_ + qbase + lo) * K_;
#pragma unroll
    for (int i = 0; i < K_; i++) {
      out_dist[base + i] = __builtin_sqrtf(fmaxf(sk[i] + n1, 0.0f));
      out_idx[base + i]  = si[i];
    }
  }
}

extern "C" void kernel_launch(void* const* d_in, const int* in_sizes, int n_in,
                              void* d_out, int out_size, void* d_ws, size_t ws_size,
                              hipStream_t stream) {
  (void)in_sizes; (void)n_in; (void)d_ws; (void)ws_size; (void)out_size;
  const float* xyz1 = (const float*)d_in[0];
  const float* xyz2 = (const float*)d_in[1];
  // d_in[2] is k (device scalar); fixed to 16 per setup_inputs.
  float* out_dist = (float*)d_out;
  int*   out_idx  = (int*)d_out + (size_t)B_ * N_ * K_;

  dim3 grid(N_ / QPB, B_);
  knn_wmma_kernel<<<grid, 256, 0, stream>>>(xyz1, xyz2, out_dist, out_idx);
}